// ONNXCompatibleNMS_88742614270444
// MI455X (gfx1250) — compile-verified
//
#include <hip/hip_runtime.h>
#include <stdint.h>

// ---------------- problem constants (match reference) ----------------
#define NMS_CONF   0.25f
#define NMS_IOU    0.45f
#define NMS_MAXOUT 100
#define NMS_BINS   1024
#define NMS_MINBIN 256     // bin of score==0.25
#define NMS_CAP    2048    // candidate capacity; power of two (bitonic sort size)
#define NMS_TARGET 1024    // gather at least this many top-score candidates
#define NMS_BLK    1024

typedef float v2f __attribute__((ext_vector_type(2)));
typedef float v8f __attribute__((ext_vector_type(8)));
typedef int   v4i __attribute__((ext_vector_type(4)));
typedef __attribute__((address_space(1))) v4i gv4i;   // global int4
typedef __attribute__((address_space(3))) v4i lv4i;   // LDS int4

#if defined(__has_builtin)
#if __has_builtin(__builtin_amdgcn_global_load_async_to_lds_b128)
#define HAVE_ASYNC_LDS 1
#endif
#endif

__global__ __launch_bounds__(NMS_BLK) void nms_single_wg_kernel(
    const float* __restrict__ boxes,   // [n,4] x1,y1,x2,y2
    const float* __restrict__ scores,  // [n]
    float* __restrict__ out,           // [100*4] boxes then [100] scores
    int n)
{
  __shared__ unsigned            sHist[NMS_BINS];   // reused as float after count
  __shared__ unsigned long long  sKeys[NMS_CAP];    // (score_bits<<32)|~idx
  __shared__ float4              sCand[NMS_CAP];    // sorted candidate boxes
  __shared__ float4              sKeptBox[NMS_MAXOUT];
  __shared__ float               sKeptScore[NMS_MAXOUT];
  __shared__ float               sKeptArea[NMS_MAXOUT];
  __shared__ float               sChunkTot[16];
  __shared__ int                 sCount, sThresh, sKept, sFlag;

  const int tid = threadIdx.x;

  // ---- 1) clear histogram / counters ----
  for (int b = tid; b < NMS_BINS; b += NMS_BLK) sHist[b] = 0u;
  if (tid == 0) { sCount = 0; sKept = 0; }
  __syncthreads();

  // ---- 2) confidence-filtered score histogram (scores in (0,1): bin map is monotone) ----
  for (int i = tid; i < n; i += NMS_BLK) {
    float s = scores[i];
    if (s > NMS_CONF) {
      int b = (int)(s * 1024.0f);
      b = b < (NMS_BINS - 1) ? b : (NMS_BINS - 1);
      atomicAdd(&sHist[b], 1u);
    }
  }
  __syncthreads();

  // ---- 3) convert histogram to float in place (counts <= 25200: exact in f32) ----
  {
    unsigned u = (tid < NMS_BINS) ? sHist[tid] : 0u;
    __syncthreads();
    if (tid < NMS_BINS) ((float*)sHist)[tid] = (float)u;
  }
  __syncthreads();

  // ---- 4) per-64-bin chunk totals via WMMA (wave 0 only; EXEC all-ones) ----
  // A = all-ones(16x4) => every D element is a column-sum of B; summing one D
  // register across 32 lanes yields 2x the sum of all 64 B inputs, independent
  // of the physical 4x16 B lane layout (layout-invariant reduction).
  if (tid < 32) {
    const float* hf = (const float*)sHist;
    v2f a; a.x = 1.0f; a.y = 1.0f;
    for (int c = 0; c < 12; ++c) {            // chunks cover bins 1023..256, top-down
      int base = 960 - 64 * c;
      v2f bv; bv.x = hf[base + tid]; bv.y = hf[base + 32 + tid];
      v8f acc = {0.f, 0.f, 0.f, 0.f, 0.f, 0.f, 0.f, 0.f};
      acc = __builtin_amdgcn_wmma_f32_16x16x4_f32(
          /*neg_a=*/false, a, /*neg_b=*/false, bv,
          /*c_mod=*/(short)0, acc, /*reuse_a=*/false, /*reuse_b=*/false);
      float v = acc[0];
      for (int m = 16; m >= 1; m >>= 1) v += __shfl_xor(v, m);
      if (tid == 0) sChunkTot[c] = 0.5f * v;  // rows 0 and 8 each hold the column sums
    }
  }
  __syncthreads();

  // ---- 5) pick threshold bin: largest T with sum_{b>=T} hist[b] >= TARGET ----
  if (tid == 0) {
    const float* hf = (const float*)sHist;
    float run = 0.0f;
    int   T = NMS_MINBIN;
    bool  found = false;
    for (int c = 0; c < 12 && !found; ++c) {
      float tot = sChunkTot[c];
      if (run + tot >= (float)NMS_TARGET) {
        int top = 1023 - 64 * c, bot = 960 - 64 * c;
        float acc = run;
        for (int b = top; b >= bot; --b) {
          acc += hf[b];
          if (acc >= (float)NMS_TARGET) { T = b; break; }
        }
        found = true;
      } else {
        run += tot;
      }
    }
    sThresh = T;
  }
  __syncthreads();

  // ---- 6) gather candidate keys above threshold ----
  {
    const int T = sThresh;
    for (int i = tid; i < n; i += NMS_BLK) {
      float s = scores[i];
      if (s > NMS_CONF) {
        int b = (int)(s * 1024.0f);
        b = b < (NMS_BINS - 1) ? b : (NMS_BINS - 1);
        if (b >= T) {
          int p = atomicAdd(&sCount, 1);
          if (p < NMS_CAP) {
            unsigned long long key =
                ((unsigned long long)__float_as_uint(s) << 32) |
                (unsigned)(~(unsigned)i);
            sKeys[p] = key;
          }
        }
      }
    }
  }
  __syncthreads();

  int count = sCount;
  count = count < NMS_CAP ? count : NMS_CAP;
  for (int p = tid; p < NMS_CAP; p += NMS_BLK)
    if (p >= count) sKeys[p] = 0ull;          // pad sorts to the front
  __syncthreads();

  // ---- 7) ascending bitonic sort of 64-bit keys; best key ends at index CAP-1 ----
  for (unsigned k = 2; k <= NMS_CAP; k <<= 1) {
    for (unsigned j = k >> 1; j > 0; j >>= 1) {
      __syncthreads();
      for (unsigned e = tid; e < NMS_CAP; e += NMS_BLK) {
        unsigned p = e ^ j;
        if (p > e) {
          bool up = ((e & k) == 0);
          unsigned long long ka = sKeys[e], kb = sKeys[p];
          if ((ka > kb) == up) { sKeys[e] = kb; sKeys[p] = ka; }
        }
      }
    }
  }
  __syncthreads();

  // ---- 8) async gather of candidate boxes into LDS, in sorted (rank) order ----
  for (int r = tid; r < NMS_CAP; r += NMS_BLK) {
    unsigned long long key = sKeys[NMS_CAP - 1 - r];
    unsigned idx = ~(unsigned)key;
    if (r >= count || idx >= (unsigned)n) idx = 0;  // harmless dummy fetch
#if defined(HAVE_ASYNC_LDS)
    __builtin_amdgcn_global_load_async_to_lds_b128(
        (gv4i*)(float*)(boxes + (size_t)idx * 4),
        (lv4i*)&sCand[r], /*offset=*/0, /*cpol=*/0);
#else
    sCand[r] = ((const float4*)boxes)[idx];
#endif
  }
#if defined(HAVE_ASYNC_LDS)
#if __has_builtin(__builtin_amdgcn_s_wait_asynccnt)
  __builtin_amdgcn_s_wait_asynccnt(0);
#else
  asm volatile("s_wait_asynccnt 0x0" ::: "memory");
#endif
#endif
  __syncthreads();

  // ---- 9) block-parallel greedy NMS (one kept box per thread) ----
  for (int r = 0; r < count; ++r) {
    __syncthreads();
    int kept = sKept;                 // uniform after barrier
    if (kept >= NMS_MAXOUT) break;
    float4 cb = sCand[r];
    bool sup = false;
    if (tid < kept) {
      float4 kb = sKeptBox[tid];
      float lx = fmaxf(cb.x, kb.x), ly = fmaxf(cb.y, kb.y);
      float rx = fminf(cb.z, kb.z), ry = fminf(cb.w, kb.w);
      float iw = fmaxf(rx - lx, 0.0f), ih = fmaxf(ry - ly, 0.0f);
      float inter = iw * ih;
      float carea = fmaxf(cb.z - cb.x, 0.0f) * fmaxf(cb.w - cb.y, 0.0f);
      float iou = inter / (sKeptArea[tid] + carea - inter + 1e-9f);
      sup = iou > NMS_IOU;
    }
    if (tid == 0) sFlag = 0;
    __syncthreads();
    if (sup) sFlag = 1;               // benign race; barrier orders it
    __syncthreads();
    if (tid == 0 && sFlag == 0) {
      sKeptBox[kept] = cb;
      sKeptArea[kept] =
          fmaxf(cb.z - cb.x, 0.0f) * fmaxf(cb.w - cb.y, 0.0f);
      unsigned long long key = sKeys[NMS_CAP - 1 - r];
      sKeptScore[kept] = __uint_as_float((unsigned)(key >> 32));
      sKept = kept + 1;
    }
  }
  __syncthreads();

  // ---- 10) write zero-padded output: [100,4] boxes then [100] scores ----
  if (tid < NMS_MAXOUT) {
    int kept = sKept;
    bool v = tid < kept;
    float4 b = v ? sKeptBox[tid] : make_float4(0.f, 0.f, 0.f, 0.f);
    out[tid * 4 + 0] = b.x;
    out[tid * 4 + 1] = b.y;
    out[tid * 4 + 2] = b.z;
    out[tid * 4 + 3] = b.w;
    out[4 * NMS_MAXOUT + tid] = v ? sKeptScore[tid] : 0.0f;
  }
}

extern "C" void kernel_launch(void* const* d_in, const int* in_sizes, int n_in,
                              void* d_out, int out_size, void* d_ws, size_t ws_size,
                              hipStream_t stream) {
  (void)n_in; (void)out_size; (void)d_ws; (void)ws_size;
  const float* boxes  = (const float*)d_in[0];
  const float* scores = (const float*)d_in[1];
  int n = in_sizes[1];               // number of boxes (scores element count)
  nms_single_wg_kernel<<<dim3(1), dim3(NMS_BLK), 0, stream>>>(
      boxes, scores, (float*)d_out, n);
}